// GCNLayer_72499047956497
// MI455X (gfx1250) — compile-verified
//
#include <hip/hip_runtime.h>
#include <hip/hip_bf16.h>

typedef __attribute__((ext_vector_type(2))) float v2f;
typedef __attribute__((ext_vector_type(8))) float v8f;

// ---------------------------------------------------------------------------
// fp32 GEMM via V_WMMA_F32_16X16X4_F32.
// C[M,N] = A[M,K] @ B[K,N], N == 256 here, K in {256,512}, M ~ 50000.
// Block = 256 threads (8 waves). Block tile = 128x64, wave tile = 16x64
// (4 accumulators of 16x16), K staged through LDS in chunks of 32.
// LDS strides padded (36 / 80 floats) for conflict-free fragment reads.
// ---------------------------------------------------------------------------
#define BM 128
#define BN 64
#define BK 32
#define LDA 36
#define LDB 80

__global__ __launch_bounds__(256) void gemm_f32_wmma(
    const float* __restrict__ A, const float* __restrict__ B,
    float* __restrict__ C, int M, int K, int N)
{
    __shared__ float As[BM * LDA];   // 18432 B
    __shared__ float Bs[BK * LDB];   // 10240 B

    const int tid  = threadIdx.x;
    const int wave = tid >> 5;       // 0..7
    const int lane = tid & 31;
    const int half = lane >> 4;      // 0 or 1
    const int ln16 = lane & 15;

    const int rowBlk = blockIdx.x * BM;
    const int colBlk = blockIdx.y * BN;

    v8f acc0 = {}, acc1 = {}, acc2 = {}, acc3 = {};

    for (int k0 = 0; k0 < K; k0 += BK) {
        // Stage A tile (BM x BK): 1024 float4 slots, 4 per thread.
#pragma unroll
        for (int it = 0; it < 4; ++it) {
            int idx = tid + it * 256;        // 0..1023
            int r   = idx >> 3;              // row in tile (8 float4 per row)
            int c4  = (idx & 7) << 2;        // float offset in row
            float4 v = {0.f, 0.f, 0.f, 0.f};
            int gr = rowBlk + r;
            if (gr < M)
                v = *reinterpret_cast<const float4*>(A + (size_t)gr * K + k0 + c4);
            *reinterpret_cast<float4*>(&As[r * LDA + c4]) = v;
        }
        // Stage B tile (BK x BN): 512 float4 slots, 2 per thread.
#pragma unroll
        for (int it = 0; it < 2; ++it) {
            int idx = tid + it * 256;        // 0..511
            int r   = idx >> 4;              // k-row (16 float4 per row)
            int c4  = (idx & 15) << 2;
            float4 v = *reinterpret_cast<const float4*>(
                B + (size_t)(k0 + r) * N + colBlk + c4);
            *reinterpret_cast<float4*>(&Bs[r * LDB + c4]) = v;
        }
        __syncthreads();

        // A fragment row for this wave: M = wave*16 + ln16.
        const float* aRow = &As[(wave * 16 + ln16) * LDA + 2 * half];
#pragma unroll
        for (int kk = 0; kk < BK; kk += 4) {
            v2f a;
            a.x = aRow[kk];
            a.y = aRow[kk + 1];
            const float* bR0 = &Bs[(kk + 2 * half)     * LDB + ln16];
            const float* bR1 = &Bs[(kk + 2 * half + 1) * LDB + ln16];
            v2f b0; b0.x = bR0[0];  b0.y = bR1[0];
            v2f b1; b1.x = bR0[16]; b1.y = bR1[16];
            v2f b2; b2.x = bR0[32]; b2.y = bR1[32];
            v2f b3; b3.x = bR0[48]; b3.y = bR1[48];
            acc0 = __builtin_amdgcn_wmma_f32_16x16x4_f32(false, a, false, b0,
                                                         (short)0, acc0, false, false);
            acc1 = __builtin_amdgcn_wmma_f32_16x16x4_f32(false, a, false, b1,
                                                         (short)0, acc1, false, false);
            acc2 = __builtin_amdgcn_wmma_f32_16x16x4_f32(false, a, false, b2,
                                                         (short)0, acc2, false, false);
            acc3 = __builtin_amdgcn_wmma_f32_16x16x4_f32(false, a, false, b3,
                                                         (short)0, acc3, false, false);
        }
        __syncthreads();
    }

    // D layout: VGPR v -> row (v + 8*half), col ln16 (+16 per tile).
    const int rbase = rowBlk + wave * 16 + 8 * half;
    const int cbase = colBlk + ln16;
#pragma unroll
    for (int e = 0; e < 8; ++e) {
        int gr = rbase + e;
        if (gr < M) {
            float* crow = C + (size_t)gr * N + cbase;
            crow[0]  = acc0[e];
            crow[16] = acc1[e];
            crow[32] = acc2[e];
            crow[48] = acc3[e];
        }
    }
}

// ---------------------------------------------------------------------------
// Edge scatter: one wave per edge. Gathers h_proj[col[e]] (L2-resident),
// scales by ew[e], atomically accumulates into accum[row[e]] (L2-resident).
// ---------------------------------------------------------------------------
__global__ __launch_bounds__(256) void edge_scatter(
    const float* __restrict__ hproj, const float* __restrict__ ew,
    const int* __restrict__ row, const int* __restrict__ col,
    float* accum, int E)
{
    const int lane = threadIdx.x & 31;
    long long e = (long long)((blockIdx.x * (size_t)blockDim.x + threadIdx.x) >> 5);
    const long long stride = (long long)((gridDim.x * (size_t)blockDim.x) >> 5);
    for (; e < E; e += stride) {
        const float w = ew[e];
        const float4* src =
            reinterpret_cast<const float4*>(hproj + (size_t)col[e] * 256);
        float* dst = accum + (size_t)row[e] * 256;
#pragma unroll
        for (int i = 0; i < 2; ++i) {
            float4 v = src[lane + 32 * i];
            int d = (lane + 32 * i) * 4;
            atomicAdd(dst + d + 0, w * v.x);
            atomicAdd(dst + d + 1, w * v.y);
            atomicAdd(dst + d + 2, w * v.z);
            atomicAdd(dst + d + 3, w * v.w);
        }
    }
}

// ---------------------------------------------------------------------------
// float4 copy (accumulator init with identity residual).
// ---------------------------------------------------------------------------
__global__ __launch_bounds__(256) void copy_f32(
    const float* __restrict__ src, float* __restrict__ dst, size_t n4)
{
    size_t i = blockIdx.x * (size_t)blockDim.x + threadIdx.x;
    const size_t stride = (size_t)gridDim.x * blockDim.x;
    const float4* s = reinterpret_cast<const float4*>(src);
    float4* d = reinterpret_cast<float4*>(dst);
    for (; i < n4; i += stride) d[i] = s[i];
}

// ---------------------------------------------------------------------------
// In-place LayerNorm over rows of 256 floats. 1 block (8 waves) per row.
// Wave-level __shfl_xor reduction + LDS cross-wave combine.
// ---------------------------------------------------------------------------
__global__ __launch_bounds__(256) void layernorm_rows(
    float* __restrict__ data, const float* __restrict__ g,
    const float* __restrict__ b, int nrows)
{
    __shared__ float red1[8];
    __shared__ float red2[8];
    const int r = blockIdx.x;
    if (r >= nrows) return;
    float* rowp = data + (size_t)r * 256;
    const int tid = threadIdx.x;

    float x = rowp[tid];

    float s = x;
#pragma unroll
    for (int off = 16; off > 0; off >>= 1) s += __shfl_xor(s, off, 32);
    if ((tid & 31) == 0) red1[tid >> 5] = s;
    __syncthreads();
    float mu = 0.f;
#pragma unroll
    for (int i = 0; i < 8; ++i) mu += red1[i];
    mu *= (1.0f / 256.0f);

    const float d = x - mu;
    float s2 = d * d;
#pragma unroll
    for (int off = 16; off > 0; off >>= 1) s2 += __shfl_xor(s2, off, 32);
    if ((tid & 31) == 0) red2[tid >> 5] = s2;
    __syncthreads();
    float var = 0.f;
#pragma unroll
    for (int i = 0; i < 8; ++i) var += red2[i];
    var *= (1.0f / 256.0f);

    rowp[tid] = d * rsqrtf(var + 1e-5f) * g[tid] + b[tid];
}

// ---------------------------------------------------------------------------
// Orchestration. Residuals are written into d_out BEFORE the edge scatter so
// the atomic accumulation lands directly on (residual + spmm); LN then runs
// in place. h_proj workspace (51.2 MB) is reused between the two edge types
// (stream ordering serializes the pipelines).
// ---------------------------------------------------------------------------
extern "C" void kernel_launch(void* const* d_in, const int* in_sizes, int n_in,
                              void* d_out, int out_size, void* d_ws, size_t ws_size,
                              hipStream_t stream)
{
    (void)n_in; (void)out_size; (void)ws_size;

    const float* x0    = (const float*)d_in[0];
    const float* x1    = (const float*)d_in[1];
    const float* W0    = (const float*)d_in[2];
    const float* W1    = (const float*)d_in[3];
    const float* Wres1 = (const float*)d_in[4];
    const float* g0    = (const float*)d_in[5];
    const float* b0    = (const float*)d_in[6];
    const float* g1    = (const float*)d_in[7];
    const float* b1    = (const float*)d_in[8];
    const float* ew0   = (const float*)d_in[9];
    const float* ew1   = (const float*)d_in[10];
    const int*   row0  = (const int*)d_in[11];
    const int*   col0  = (const int*)d_in[12];
    const int*   row1  = (const int*)d_in[13];
    const int*   col1  = (const int*)d_in[14];

    const int D0 = 256, D1 = 512, DO = 256;
    const int N0 = in_sizes[0] / D0;
    const int N1 = in_sizes[1] / D1;
    const int E0 = in_sizes[9];
    const int E1 = in_sizes[10];

    float* hproj = (float*)d_ws;                    // N*DO floats, reused
    float* out0  = (float*)d_out;                   // rows [0, N0)
    float* out1  = out0 + (size_t)N0 * DO;          // rows [N0, N0+N1)

    const dim3 blk(256);

    // ---- type 0: identity residual ----
    dim3 gg0((N0 + BM - 1) / BM, DO / BN);
    gemm_f32_wmma<<<gg0, blk, 0, stream>>>(x0, W0, hproj, N0, D0, DO);
    copy_f32<<<2048, blk, 0, stream>>>(x0, out0, (size_t)N0 * DO / 4);
    edge_scatter<<<(E0 + 7) / 8, blk, 0, stream>>>(hproj, ew0, row0, col0, out0, E0);
    layernorm_rows<<<N0, blk, 0, stream>>>(out0, g0, b0, N0);

    // ---- type 1: projected residual ----
    dim3 gg1((N1 + BM - 1) / BM, DO / BN);
    gemm_f32_wmma<<<gg1, blk, 0, stream>>>(x1, Wres1, out1, N1, D1, DO);
    gemm_f32_wmma<<<gg1, blk, 0, stream>>>(x1, W1, hproj, N1, D1, DO);
    edge_scatter<<<(E1 + 7) / 8, blk, 0, stream>>>(hproj, ew1, row1, col1, out1, E1);
    layernorm_rows<<<N1, blk, 0, stream>>>(out1, g1, b1, N1);
}